// Interaction_Transition_Model_40501541601733
// MI455X (gfx1250) — compile-verified
//
#include <hip/hip_runtime.h>
#include <math.h>

#define N_ROWS 8192
#define IN_CH  128
#define GW     64
#define MLP_H  256
#define WHEELBASE 2.96f
#define DT     0.2f
#define C_R    0.1f
#define C_A    0.5f
#define LN_EPS 1e-5f
#define MAX_STEER 1.0471975511965976f  /* deg2rad(60) */

typedef __attribute__((ext_vector_type(2))) float v2f;
typedef __attribute__((ext_vector_type(8))) float v8f;
typedef int v4i __attribute__((vector_size(16)));   // matches builtin param type

#define AS1 __attribute__((address_space(1)))
#define AS3 __attribute__((address_space(3)))

#if __has_builtin(__builtin_amdgcn_global_load_async_to_lds_b128)
#define HAVE_ASYNC_LDS 1
#else
#define HAVE_ASYNC_LDS 0
#endif

// ---------------------------------------------------------------------------
// Kernel 1: K = obs@Wk+bk and V = obs@Wv+bv via V_WMMA_F32_16X16X4_F32.
//
// One 256-thread block (8 waves) per 16-row obs slab. The slab (16x128 f32,
// 8 KB) is staged into LDS once per block -- with GLOBAL_LOAD_ASYNC_TO_LDS
// (ASYNCcnt path) when the builtin is available -- then the 8 waves each
// produce one 16x16 output tile: wave = {K|V} x {4 N-tiles}. This removes
// the 8x redundant global reads of obs that a wave-per-tile scheme incurs.
//
// Fragment layouts (ISA 7.12.2, f32 16x16x4):
//   A (16x4): lanes 0-15 hold {K=0,K=1}, lanes 16-31 hold {K=2,K=3}
//   B (4x16): mirrored striping over N
//   C/D (16x16, 8 VGPRs): row = r + 8*(lane>=16), col = lane&15
// ---------------------------------------------------------------------------
__global__ void __launch_bounds__(256) kv_proj_wmma(
    const float* __restrict__ obs,
    const float* __restrict__ Wk, const float* __restrict__ bk,
    const float* __restrict__ Wv, const float* __restrict__ bv,
    float* __restrict__ Kbuf, float* __restrict__ Vbuf)
{
    __shared__ float tileA[16 * IN_CH];          // 8 KB: rows m0..m0+15 of obs

    const int m0   = blockIdx.x * 16;
    const int base = m0 * IN_CH;                 // tile is contiguous in obs

    // ---- stage obs slab into LDS (512 float4s, 2 per thread) ----
#if HAVE_ASYNC_LDS
    for (int i = threadIdx.x; i < 512; i += 256) {
        __builtin_amdgcn_global_load_async_to_lds_b128(
            (AS1 v4i*)(obs + base + i * 4),
            (AS3 v4i*)(&tileA[i * 4]),
            /*offset=*/0, /*cpol=*/0);
    }
#if __has_builtin(__builtin_amdgcn_s_wait_asynccnt)
    __builtin_amdgcn_s_wait_asynccnt(0);
#else
    asm volatile("s_wait_asynccnt 0x0" ::: "memory");
#endif
#else
    for (int i = threadIdx.x; i < 512; i += 256) {
        *(float4*)&tileA[i * 4] = *(const float4*)(obs + base + i * 4);
    }
#endif
    __syncthreads();

    // ---- per-wave tile assignment ----
    const int wave = threadIdx.x >> 5;
    const int lane = threadIdx.x & 31;
    const int mat  = wave >> 2;                  // 0 = K, 1 = V
    const int n0   = (wave & 3) * 16;

    const float* __restrict__ W    = mat ? Wv : Wk;
    const float* __restrict__ bias = mat ? bv : bk;
    float* __restrict__ out        = mat ? Vbuf : Kbuf;

    const int half = lane >> 4;                  // 0: lanes 0-15, 1: 16-31
    const int lid  = lane & 15;
    const int koff = half * 2;                   // K offset inside 4-wide slab
    const int bcol = n0 + lid;                   // output column of this lane

    v8f acc = {};
    #pragma unroll
    for (int k0 = 0; k0 < IN_CH; k0 += 4) {
        v2f a;                                   // A from LDS (ds_load_b64)
        a.x = tileA[lid * IN_CH + k0 + koff];
        a.y = tileA[lid * IN_CH + k0 + koff + 1];
        v2f b;                                   // B from global (L2-resident)
        const float* wp = W + (k0 + koff) * GW + bcol;
        if (k0 + 8 < IN_CH) __builtin_prefetch(wp + 8 * GW, 0, 0);
        b.x = wp[0];
        b.y = wp[GW];
        // 8 args: (neg_a, A, neg_b, B, c_mod, C, reuse_a, reuse_b)
        acc = __builtin_amdgcn_wmma_f32_16x16x4_f32(false, a, false, b,
                                                    (short)0, acc, false, false);
    }
    const float bn = bias[bcol];
    #pragma unroll
    for (int r = 0; r < 8; ++r) {
        const int row = m0 + r + half * 8;
        out[row * GW + bcol] = acc[r] + bn;
    }
}

// ---------------------------------------------------------------------------
// Kernel 2: q0[n] = obs[0,:] @ Wq[:,n] + bq[n]   (64 outputs)
// ---------------------------------------------------------------------------
__global__ void __launch_bounds__(64) q0_kernel(
    const float* __restrict__ obs, const float* __restrict__ Wq,
    const float* __restrict__ bq, float* __restrict__ q0)
{
    const int n = threadIdx.x;
    float acc = bq[n];
    #pragma unroll 8
    for (int c = 0; c < IN_CH; ++c)
        acc = fmaf(obs[c], Wq[c * GW + n], acc);
    q0[n] = acc;
}

// ---------------------------------------------------------------------------
// Kernel 3a: s[j] = q0 . K[j,:]   for j in [0,8192)
// ---------------------------------------------------------------------------
__global__ void __launch_bounds__(256) scores_kernel(
    const float* __restrict__ Kbuf, const float* __restrict__ q0,
    float* __restrict__ s)
{
    __shared__ float q[GW];
    if (threadIdx.x < GW) q[threadIdx.x] = q0[threadIdx.x];
    __syncthreads();
    const int j = blockIdx.x * blockDim.x + threadIdx.x;
    const float* kr = Kbuf + j * GW;
    float acc = 0.0f;
    #pragma unroll
    for (int c = 0; c < GW; ++c) acc = fmaf(q[c], kr[c], acc);
    s[j] = acc;
}

// ---------------------------------------------------------------------------
// Kernel 3b: softmax over s[0..8191], then out0[n] = sum_j p_j * V[j,n].
// Single workgroup; block-tree reductions in LDS.
// ---------------------------------------------------------------------------
__global__ void __launch_bounds__(256) softmax_av_kernel(
    float* __restrict__ s, const float* __restrict__ Vbuf,
    float* __restrict__ out0)
{
    __shared__ float red[256];
    const int t = threadIdx.x;

    // global max
    float m = -3.402823466e38f;
    for (int j = t; j < N_ROWS; j += 256) m = fmaxf(m, s[j]);
    red[t] = m; __syncthreads();
    for (int off = 128; off > 0; off >>= 1) {
        if (t < off) red[t] = fmaxf(red[t], red[t + off]);
        __syncthreads();
    }
    m = red[0]; __syncthreads();

    // exp + sum (overwrite s with unnormalized p)
    float ls = 0.0f;
    for (int j = t; j < N_ROWS; j += 256) {
        const float p = expf(s[j] - m);
        s[j] = p;
        ls += p;
    }
    red[t] = ls; __syncthreads();
    for (int off = 128; off > 0; off >>= 1) {
        if (t < off) red[t] += red[t + off];
        __syncthreads();
    }
    const float invZ = 1.0f / red[0];
    __syncthreads();

    // out0[n] = invZ * sum_j p_j * V[j,n]; 4 j-stripes x 64 columns
    const int n = t & (GW - 1);
    const int stripe = t >> 6;                 // 0..3
    const int jb = stripe * (N_ROWS / 4);
    const int je = jb + (N_ROWS / 4);
    float acc = 0.0f;
    for (int j = jb; j < je; ++j)
        acc = fmaf(s[j], Vbuf[j * GW + n], acc);
    red[t] = acc; __syncthreads();
    if (t < GW)
        out0[n] = (red[n] + red[n + 64] + red[n + 128] + red[n + 192]) * invZ;
}

// ---------------------------------------------------------------------------
// Kernel 4: h0=[out0, action[0,:]] -> W1+b1 -> LayerNorm -> ReLU -> W2+b2
// Produces throttle (ta[0]) and delta (ta[1]). Single 256-thread block.
// ---------------------------------------------------------------------------
__global__ void __launch_bounds__(256) head_kernel(
    const float* __restrict__ out0, const float* __restrict__ action,
    const float* __restrict__ W1, const float* __restrict__ b1,
    const float* __restrict__ ln_g, const float* __restrict__ ln_b,
    const float* __restrict__ W2, const float* __restrict__ b2,
    float* __restrict__ ta)
{
    __shared__ float h[GW + 2];
    __shared__ float red[256];
    const int t = threadIdx.x;
    if (t < GW) h[t] = out0[t];
    if (t < 2)  h[GW + t] = action[t];   // act4 row 0 == action row 0
    __syncthreads();

    float z = b1[t];
    #pragma unroll
    for (int i = 0; i < GW + 2; ++i)
        z = fmaf(h[i], W1[i * MLP_H + t], z);

    // mean
    red[t] = z; __syncthreads();
    for (int off = 128; off > 0; off >>= 1) {
        if (t < off) red[t] += red[t + off];
        __syncthreads();
    }
    const float mu = red[0] * (1.0f / MLP_H);
    __syncthreads();

    // variance
    const float d = z - mu;
    red[t] = d * d; __syncthreads();
    for (int off = 128; off > 0; off >>= 1) {
        if (t < off) red[t] += red[t + off];
        __syncthreads();
    }
    const float var = red[0] * (1.0f / MLP_H);
    __syncthreads();

    float zn = d * rsqrtf(var + LN_EPS) * ln_g[t] + ln_b[t];
    zn = fmaxf(zn, 0.0f);

    // pred_action[0,c] = zn . W2[:,c] + b2[c]
    red[t] = zn * W2[t * 2 + 0]; __syncthreads();
    for (int off = 128; off > 0; off >>= 1) {
        if (t < off) red[t] += red[t + off];
        __syncthreads();
    }
    if (t == 0) ta[0] = red[0] + b2[0];
    __syncthreads();
    red[t] = zn * W2[t * 2 + 1]; __syncthreads();
    for (int off = 128; off > 0; off >>= 1) {
        if (t < off) red[t] += red[t + off];
        __syncthreads();
    }
    if (t == 0) ta[1] = red[0] + b2[1];
}

// ---------------------------------------------------------------------------
// Kernel 5: kinematic bicycle model, elementwise over 8192 rows.
// ---------------------------------------------------------------------------
__global__ void __launch_bounds__(256) dynamics_kernel(
    const float* __restrict__ obs, const float* __restrict__ ta,
    float* __restrict__ out)
{
    const int j = blockIdx.x * blockDim.x + threadIdx.x;
    if (j >= N_ROWS) return;
    const float throttle = ta[0];
    const float delta    = ta[1];
    const float* o = obs + j * IN_CH;
    const float x = o[0], y = o[1], vx = o[2], vy = o[3], yaw = o[4];

    const float v0 = sqrtf(vx * vx + vy * vy);
    const float f_load = v0 * (C_R + C_A * v0);
    const float v1 = v0 + DT * (throttle - f_load);
    const float dd = fminf(fmaxf(delta, -MAX_STEER), MAX_STEER);
    const float omega = v1 * tanf(dd) / WHEELBASE;
    const float x1 = x + v1 * cosf(yaw) * DT;
    const float y1 = y + v1 * sinf(yaw) * DT;
    const float a  = yaw + omega * DT;
    const float yaw1 = atan2f(sinf(a), cosf(a));
    const float cy = cosf(yaw1), sy = sinf(yaw1);

    float* po = out + j * 5;
    po[0] = x1;
    po[1] = y1;
    po[2] = v1 * cy;
    po[3] = v1 * sy;
    po[4] = yaw1;
}

// ---------------------------------------------------------------------------
extern "C" void kernel_launch(void* const* d_in, const int* in_sizes, int n_in,
                              void* d_out, int out_size, void* d_ws, size_t ws_size,
                              hipStream_t stream) {
    (void)in_sizes; (void)n_in; (void)out_size; (void)ws_size;
    const float* obs    = (const float*)d_in[0];
    const float* action = (const float*)d_in[1];
    const float* Wq     = (const float*)d_in[2];
    const float* bq     = (const float*)d_in[3];
    const float* Wk     = (const float*)d_in[4];
    const float* bk     = (const float*)d_in[5];
    const float* Wv     = (const float*)d_in[6];
    const float* bv     = (const float*)d_in[7];
    const float* W1     = (const float*)d_in[8];
    const float* b1     = (const float*)d_in[9];
    const float* ln_g   = (const float*)d_in[10];
    const float* ln_b   = (const float*)d_in[11];
    const float* W2     = (const float*)d_in[12];
    const float* b2     = (const float*)d_in[13];

    float* ws   = (float*)d_ws;
    float* Kbuf = ws;                       // 8192*64
    float* Vbuf = Kbuf + N_ROWS * GW;       // 8192*64
    float* q0   = Vbuf + N_ROWS * GW;       // 64
    float* s    = q0 + GW;                  // 8192
    float* out0 = s + N_ROWS;               // 64
    float* ta   = out0 + GW;                // 2
    float* outp = (float*)d_out;            // (8192,5) f32

    kv_proj_wmma<<<N_ROWS / 16, 256, 0, stream>>>(obs, Wk, bk, Wv, bv, Kbuf, Vbuf);
    q0_kernel<<<1, 64, 0, stream>>>(obs, Wq, bq, q0);
    scores_kernel<<<N_ROWS / 256, 256, 0, stream>>>(Kbuf, q0, s);
    softmax_av_kernel<<<1, 256, 0, stream>>>(s, Vbuf, out0);
    head_kernel<<<1, 256, 0, stream>>>(out0, action, W1, b1, ln_g, ln_b, W2, b2, ta);
    dynamics_kernel<<<N_ROWS / 256, 256, 0, stream>>>(obs, ta, outp);
}